// SDEModel_66108136620070
// MI455X (gfx1250) — compile-verified
//
#include <hip/hip_runtime.h>
#include <stdint.h>

#define TPB 32   // trajectories per block == one wave32 per workgroup
#define TC  50   // time steps per staged chunk (2000 % 50 == 0)

typedef uint32_t u32;
typedef uint32_t u32v4 __attribute__((ext_vector_type(4)));
typedef uint32_t u32v8 __attribute__((ext_vector_type(8)));

__device__ __forceinline__ float tanh_fast(float v) {
#if __has_builtin(__builtin_amdgcn_tanhf)
  return __builtin_amdgcn_tanhf(v);        // v_tanh_f32 (CDNA5 trans op)
#else
  float r;
  asm("v_tanh_f32 %0, %1" : "=v"(r) : "v"(v));
  return r;
#endif
}

// softplus(u) + 1e-4, numerically stable: max(u,0) + log(1 + exp(-|u|))
__device__ __forceinline__ float softplus_eps(float u) {
  float m = fmaxf(u, 0.0f);
  float e = __expf(-fabsf(u));             // v_exp_f32
  return m + __logf(1.0f + e) + 1e-4f;     // v_log_f32
}

struct Net { float wx[4], wt[4], b1[4], w2[4], b2; };

__device__ __forceinline__ float mlp_sp(const Net& n, float x, float t) {
  float h0 = tanh_fast(fmaf(x, n.wx[0], fmaf(t, n.wt[0], n.b1[0])));
  float h1 = tanh_fast(fmaf(x, n.wx[1], fmaf(t, n.wt[1], n.b1[1])));
  float h2 = tanh_fast(fmaf(x, n.wx[2], fmaf(t, n.wt[2], n.b1[2])));
  float h3 = tanh_fast(fmaf(x, n.wx[3], fmaf(t, n.wt[3], n.b1[3])));
  float u = n.b2;
  u = fmaf(h0, n.w2[0], u);
  u = fmaf(h1, n.w2[1], u);
  u = fmaf(h2, n.w2[2], u);
  u = fmaf(h3, n.w2[3], u);
  return softplus_eps(u);
}

// Issue one TDM 2D tile load: rows x n_cols fp32 elements, row stride
// stride_elems, global -> LDS at byte offset lds_off. All operands uniform;
// tensor ops are wave-level and EXEC-independent.
__device__ __forceinline__ void tdm_issue(const float* gsrc, u32 lds_off,
                                          int n_cols, int rows, int stride_elems) {
  uint64_t ga = (uint64_t)(uintptr_t)gsrc;
  u32v4 g0;
  g0.x = 1u;                                             // count=1 (valid), user mode
  g0.y = lds_off;                                        // lds_addr (bytes)
  g0.z = (u32)ga;                                        // global_addr[31:0]
  g0.w = ((u32)(ga >> 32) & 0x01FFFFFFu) | 0x80000000u;  // addr[56:32] | type=2
  u32v8 g1;
  g1.s0 = 2u << 16;                                      // data_size=4B; no mask/pad/iter
  g1.s1 = ((u32)stride_elems & 0xFFFFu) << 16;           // tensor_dim0[15:0]
  g1.s2 = (((u32)stride_elems >> 16) & 0xFFFFu)          // tensor_dim0[31:16]
        | (((u32)rows & 0xFFFFu) << 16);                 // tensor_dim1[15:0]
  g1.s3 = ((u32)n_cols & 0xFFFFu) << 16;                 // tensor_dim1 hi=0 | tile_dim0
  g1.s4 = (u32)rows & 0xFFFFu;                           // tile_dim1 ; tile_dim2=0
  g1.s5 = (u32)stride_elems;                             // tensor_dim0_stride[31:0]
  g1.s6 = 0u;                                            // stride hi ; dim1_stride lo
  g1.s7 = 0u;
  asm volatile("tensor_load_to_lds %0, %1" :: "s"(g0), "s"(g1) : "memory");
}

__global__ void __launch_bounds__(TPB)
sde_em_kernel(const float* __restrict__ ts, const float* __restrict__ x0,
              const float* __restrict__ noise,
              const float* __restrict__ Wf1, const float* __restrict__ bf1,
              const float* __restrict__ Wf2, const float* __restrict__ bf2,
              const float* __restrict__ Wg1, const float* __restrict__ bg1,
              const float* __restrict__ Wg2, const float* __restrict__ bg2,
              float* __restrict__ out, int N, int T) {
  __shared__ float zbuf[2][TC][TPB];       // double-buffered noise tiles (TDM dest)
  __shared__ float xstage[TPB][TC + 1];    // padded: stride 51 coprime with 64 banks

  const int lane = threadIdx.x;            // one wave32 per block
  const int trajBase = blockIdx.x * TPB;
  const int traj = trajBase + lane;

  Net nf, ng;
#pragma unroll
  for (int k = 0; k < 4; ++k) {
    nf.wx[k] = Wf1[k];  nf.wt[k] = Wf1[4 + k];  nf.b1[k] = bf1[k];  nf.w2[k] = Wf2[k];
    ng.wx[k] = Wg1[k];  ng.wt[k] = Wg1[4 + k];  ng.b1[k] = bg1[k];  ng.w2[k] = Wg2[k];
  }
  nf.b2 = bf2[0];  ng.b2 = bg2[0];

  float x = x0[0];
  out[(size_t)traj * T] = x;               // t = 0 column

  float t = ts[0];
  const int steps  = T - 1;
  const int nchunk = steps / TC;

  // Prologue: prefetch chunks 0 and 1 (one TDM op each).
  {
    const float* base = noise + (size_t)trajBase;
    if (nchunk > 0)
      tdm_issue(base, (u32)(uintptr_t)&zbuf[0][0][0], TPB, TC, N);
    if (nchunk > 1)
      tdm_issue(base + (size_t)TC * N, (u32)(uintptr_t)&zbuf[1][0][0], TPB, TC, N);
  }

  int step = 0;
  for (int c = 0; c < nchunk; ++c) {
    if (c + 1 < nchunk) __builtin_amdgcn_s_wait_tensorcnt(1);  // chunk c resident,
    else                __builtin_amdgcn_s_wait_tensorcnt(0);  // c+1 stays in flight
    __syncthreads();                       // LDS fence (single-wave: near-free)

    const float* zs = &zbuf[c & 1][0][0];
#pragma unroll 2
    for (int s = 0; s < TC; ++s, ++step) {
      float z  = zs[s * TPB + lane];       // ds_load, bank-conflict-free
      float tn = ts[step + 1];             // uniform, cache-broadcast
      float dt = tn - t;
      float f  = mlp_sp(nf, x, t);
      float g  = mlp_sp(ng, x, t);
      x = fmaf(f, dt, fmaf(g * sqrtf(dt), z, x));
      xstage[lane][s] = x;
      t = tn;
    }
    __syncthreads();                       // xstage readable across lanes

    if (c + 2 < nchunk)                    // prefetch chunk c+2 into freed buffer
      tdm_issue(noise + (size_t)(c + 2) * TC * N + (size_t)trajBase,
                (u32)(uintptr_t)&zbuf[c & 1][0][0], TPB, TC, N);

    // Coalesced transposed write-out: the wave stores each trajectory's
    // TC contiguous time values -> 128B bursts into out[traj*T + ...].
    const size_t tbase = (size_t)c * TC + 1;
    for (int r = 0; r < TPB; ++r) {
      float* orow = out + (size_t)(trajBase + r) * T + tbase;
      for (int s0 = lane; s0 < TC; s0 += 32)
        orow[s0] = xstage[r][s0];
    }
    __syncthreads();                       // xstage free before next chunk
  }

  // Tail (steps % TC != 0): direct path (empty for T=2001, TC=50).
  for (; step < steps; ++step) {
    float z  = noise[(size_t)step * N + traj];
    float tn = ts[step + 1];
    float dt = tn - t;
    float f  = mlp_sp(nf, x, t);
    float g  = mlp_sp(ng, x, t);
    x = fmaf(f, dt, fmaf(g * sqrtf(dt), z, x));
    out[(size_t)traj * T + step + 1] = x;
    t = tn;
  }
}

extern "C" void kernel_launch(void* const* d_in, const int* in_sizes, int n_in,
                              void* d_out, int out_size, void* d_ws, size_t ws_size,
                              hipStream_t stream) {
  (void)n_in; (void)d_ws; (void)ws_size; (void)out_size;
  const float* ts  = (const float*)d_in[0];
  const float* x0  = (const float*)d_in[1];
  const float* nz  = (const float*)d_in[2];
  const float* Wf1 = (const float*)d_in[3];
  const float* bf1 = (const float*)d_in[4];
  const float* Wf2 = (const float*)d_in[5];
  const float* bf2 = (const float*)d_in[6];
  const float* Wg1 = (const float*)d_in[7];
  const float* bg1 = (const float*)d_in[8];
  const float* Wg2 = (const float*)d_in[9];
  const float* bg2 = (const float*)d_in[10];
  float* out = (float*)d_out;

  const int T = in_sizes[0];                       // 2001
  const int N = in_sizes[2] / (T - 1);             // 16384

  dim3 grid(N / TPB), block(TPB);                  // 512 single-wave blocks:
  sde_em_kernel<<<grid, block, 0, stream>>>(       // max WGP spread for a
      ts, x0, nz, Wf1, bf1, Wf2, bf2, Wg1, bg1,    // parallelism-starved scan
      Wg2, bg2, out, N, T);
}